// GraphLoss_58660663329229
// MI455X (gfx1250) — compile-verified
//
#include <hip/hip_runtime.h>

// GraphLoss on MI455X (gfx1250, wave32).
// Memory-bound streaming kernel (~35.6 MB total -> ~1.5us at 23.3 TB/s).
// The only surviving matmul (16-state x 15-feature logits) runs on the
// matrix pipe via V_WMMA_F32_16X16X4_F32 (fp32-exact).

#define TOTAL   15
#define NSHAPE  29
#define NATT    96
#define SSTRIDE 44    // TOTAL + NSHAPE
#define ASTRIDE 111   // TOTAL + NATT

typedef float v2f __attribute__((ext_vector_type(2)));
typedef float v8f __attribute__((ext_vector_type(8)));

// 15-bit column mask of state-space row `st` (which feature columns are 1).
// Row 0 is all-zero; rows 1..15 are {parent} and {parent,leaf} per HIERARCHY.
__device__ __forceinline__ unsigned state_mask(int st) {
  if (st == 0) return 0u;
  int g = (st - 1) / 5;      // group: parent 12+g
  int r = (st - 1) % 5;      // 0 = parent-only, 1..4 = leaf 4g+r-1
  unsigned m = 1u << (12 + g);
  if (r) m |= 1u << (4 * g + (r - 1));
  return m;
}

// 16-bit mask over STATES selected by label L (bit st = ind[st, L] > 0).
// Leaf L<12 -> single state 2 + 5*(L/4) + (L%4); parent -> 5 states 0x3E<<5g.
__device__ __forceinline__ unsigned sel_states(int L) {
  return (L < 12) ? (1u << (2 + (L >> 2) * 5 + (L & 3)))
                  : (0x3Eu << ((L - 12) * 5));
}

// D = S(16x15, 0/1) * X^T(15x16 samples) as a 16x16 fp32 tile.
// 4 chained V_WMMA_F32_16X16X4_F32, K padded 15->16 with zeros.
// Wave32 operand layout: lane = (n | hi<<4). A: lane<16 holds (M=n, K=k..k+1),
// lane>=16 holds (M=n, K=k+2..k+3); B mirrors with N=n columns.
// The constant A operand (aop) is synthesized once by the caller and reused
// across all three feature blocks.
__device__ __forceinline__ v8f logits16(const float* __restrict__ xrow,
                                        const v2f aop[4], int hi) {
  v8f c = {0.f, 0.f, 0.f, 0.f, 0.f, 0.f, 0.f, 0.f};
#pragma unroll
  for (int s = 0; s < 4; ++s) {
    const int k = 4 * s + 2 * hi;   // k <= 14, always in-bounds
    const int f1 = k + 1;           // f1 == 15 only for (s=3, hi=1): zero pad
    v2f b;
    b.x = xrow[k];
    // branchless pad: clamp index, zero weight (avoids exec-mask branches)
    const int   f1c = (f1 < TOTAL) ? f1 : 0;
    const float w1  = (f1 < TOTAL) ? 1.f : 0.f;
    b.y = xrow[f1c] * w1;
    c = __builtin_amdgcn_wmma_f32_16x16x4_f32(false, aop[s], false, b,
                                              (short)0, c, false, false);
  }
  return c;
}

__global__ void zero_out3(float* out) {
  if (threadIdx.x < 3) out[threadIdx.x] = 0.0f;
}

__global__ void __launch_bounds__(256)
graph_loss_kernel(const float* __restrict__ fs,
                  const int*   __restrict__ labels,
                  const float* __restrict__ label_fs,
                  const int*   __restrict__ labels_shape,
                  const int*   __restrict__ att_label,
                  const float* __restrict__ att_fs,
                  float* __restrict__ out, int Btot) {
  const int lane = threadIdx.x & 31;
  const int wave = threadIdx.x >> 5;
  const int n    = lane & 15;     // sample-in-tile / state row for A operand
  const int hi   = lane >> 4;     // K-half selector
  const unsigned amask = state_mask(n);

  // Synthesize the constant S operand once (reused by all three WMMA chains).
  v2f aop[4];
#pragma unroll
  for (int s = 0; s < 4; ++s) {
    const int k = 4 * s + 2 * hi;
    aop[s].x = (float)((amask >> k) & 1u);
    aop[s].y = (float)((amask >> (k + 1)) & 1u);  // bit 15 is 0 (pad)
  }

  const int gw = blockIdx.x * (blockDim.x >> 5) + wave;  // one 16-sample tile/wave
  const int s0 = gw * 16;

  float l1 = 0.f, l2 = 0.f, l3 = 0.f;

  if (s0 < Btot) {               // wave-uniform branch: EXEC stays all-ones inside
    int row = s0 + n;
    const float val = (row < Btot) ? 1.f : 0.f;
    if (row >= Btot) row = Btot - 1;

    // ---------- part 1: hierarchy loss ----------
    const float* r1 = fs + (size_t)row * TOTAL;
    v8f c1 = logits16(r1, aop, hi);
    const unsigned selh = sel_states(labels[row]) >> (8 * hi);
    float zp = 0.f, np = 0.f;
#pragma unroll
    for (int i = 0; i < 8; ++i) {
      float e = __expf(c1[i]);                 // D: lane<16 -> M=i, lane>=16 -> M=8+i
      zp += e;
      np += ((selh >> i) & 1u) ? e : 0.f;      // constant bit test -> cndmask
    }
    zp += __shfl_xor(zp, 16);
    np += __shfl_xor(np, 16);
    const float p = np / zp;                   // z includes row0's exp(0)=1
    l1 = -__logf(p) * val;

    // ---------- part 2: shape loss ----------
    // z_s = 1 + A * sum_k e^{b_k};  pt = A * e^{b_lab} / z_s
    const float* r2 = label_fs + (size_t)row * SSTRIDE;
    v8f c2 = logits16(r2, aop, hi);
    float Ap = 0.f;
#pragma unroll
    for (int i = 0; i < 8; ++i) Ap += __expf(c2[i]);
    float A2 = Ap + __shfl_xor(Ap, 16);
    float bs = 0.f;
    for (int k = hi; k < NSHAPE; k += 2) bs += __expf(r2[TOTAL + k]);
    bs += __shfl_xor(bs, 16);
    const int lab = labels_shape[row];         // in [15, 44)
    const float eb = __expf(r2[lab]);
    const float zs = 1.0f + A2 * bs;
    const float pt = A2 * eb / zs;
    const float om1 = 1.0f - p;
    l2 = -(om1 * om1) * __logf(pt) * val;

    // ---------- part 3: attribute loss ----------
    // z_a = 1 + A * sum_k e^{c_k};  num = A * sum_k [pos_k] e^{c_k}
    const float* r3 = att_fs + (size_t)row * ASTRIDE;
    const int* al = att_label + (size_t)row * NATT;
    v8f c3 = logits16(r3, aop, hi);
    float Aap = 0.f;
#pragma unroll
    for (int i = 0; i < 8; ++i) Aap += __expf(c3[i]);
    float A3 = Aap + __shfl_xor(Aap, 16);
    float Cs = 0.f, Cp = 0.f;
    for (int k = hi; k < NATT; k += 2) {
      float e = __expf(r3[TOTAL + k]);
      Cs += e;
      Cp += (float)al[k] * e;                  // labels are 0/1: branchless fma
    }
    Cs += __shfl_xor(Cs, 16);
    Cp += __shfl_xor(Cp, 16);
    const float za  = 1.0f + A3 * Cs;
    const float pa  = A3 * Cp / za;
    const float om2 = 1.0f - pt;
    l3 = -(om2 * om2) * __logf(pa) * val;
  }

  // ---------- reduction: wave butterfly -> LDS -> one atomicAdd per block ----
  // Every sample's losses are duplicated in both lane halves -> scale by 0.5.
#pragma unroll
  for (int off = 16; off > 0; off >>= 1) {
    l1 += __shfl_xor(l1, off);
    l2 += __shfl_xor(l2, off);
    l3 += __shfl_xor(l3, off);
  }
  __shared__ float red[3][8];
  if (lane == 0) { red[0][wave] = l1; red[1][wave] = l2; red[2][wave] = l3; }
  __syncthreads();
  if (threadIdx.x == 0) {
    float t1 = 0.f, t2 = 0.f, t3 = 0.f;
#pragma unroll
    for (int w = 0; w < 8; ++w) { t1 += red[0][w]; t2 += red[1][w]; t3 += red[2][w]; }
    const float sc = 0.5f / (float)Btot;   // 0.5 undoes lane-half duplication
    atomicAdd(out + 0, t1 * sc);
    atomicAdd(out + 1, t2 * sc);
    atomicAdd(out + 2, t3 * sc);
  }
}

extern "C" void kernel_launch(void* const* d_in, const int* in_sizes, int n_in,
                              void* d_out, int out_size, void* d_ws, size_t ws_size,
                              hipStream_t stream) {
  const float* fs           = (const float*)d_in[0];
  const int*   labels       = (const int*)  d_in[1];
  const float* label_fs     = (const float*)d_in[2];
  const int*   labels_shape = (const int*)  d_in[3];
  const int*   att_label    = (const int*)  d_in[4];
  const float* att_fs       = (const float*)d_in[5];
  // d_in[6..8] (state spaces) are deterministic structure; baked in as constants.
  float* out = (float*)d_out;
  const int B = in_sizes[1];                     // labels: one per sample

  hipLaunchKernelGGL(zero_out3, dim3(1), dim3(32), 0, stream, out);

  const int tiles  = (B + 15) / 16;              // 16 samples per wave
  const int wpb    = 256 / 32;                   // 8 waves per block
  const int blocks = (tiles + wpb - 1) / wpb;
  hipLaunchKernelGGL(graph_loss_kernel, dim3(blocks), dim3(256), 0, stream,
                     fs, labels, label_fs, labels_shape, att_label, att_fs,
                     out, B);
}